// SampleAndGroupModule_77257871721216
// MI455X (gfx1250) — compile-verified
//
#include <hip/hip_runtime.h>
#include <hip/hip_bf16.h>
#include <math.h>

// ---------------------------------------------------------------------------
// SampleAndGroup + 2-layer MLP (BN+ReLU) + group max-pool for MI455X (gfx1250)
// Matrix ops: V_WMMA_F32_16X16X4_F32 (wave32, fp32-exact vs reference).
// GEMM2 stages tiles via GLOBAL_LOAD_ASYNC_TO_LDS_B128 + ASYNCcnt double-buffer.
// ---------------------------------------------------------------------------

typedef __attribute__((ext_vector_type(2))) float v2f;
typedef __attribute__((ext_vector_type(8))) float v8f;

#define N_PTS   65536
#define NSAMP   2048
#define NGRP    64
#define EMB     512
#define NROWS   (NSAMP * NGRP)   // 131072
#define FEAT_K  12
#define BN_EPS  1e-5f

// LDS tile pitches (floats). Both give 16B-aligned rows (async B128 dest
// requires 16B alignment) and conflict-free fragment reads:
//  A: 20*4=80B per row;  20*row mod 64 distinct for 16 rows.
//  B: 72*4=288B per row; 72 mod 64 = 8 -> the two half-wave groups read
//     disjoint bank ranges.
#define APITCH 20
#define BPITCH 72

// ---- workspace layout (bytes) ----
#define OFF_SAMPLED 0                         // int[2048]
#define OFF_DISTS   8192                      // float[65536]
#define OFF_PNORM   270336                    // float[65536]
#define OFF_GROUPS  532480                    // int[131072]
#define OFF_FEAT    1056768                   // float[131072*12]
#define OFF_SUM     7348224                   // float[512]
#define OFF_SUMSQ   7350272                   // float[512]
#define OFF_MEAN    7352320                   // float[512]
#define OFF_RSTD    7354368                   // float[512]
#define OFF_H1      7356416                   // float[131072*512]
#define OFF_H2      275791872                 // float[131072*512]

// ---- ASYNCcnt wait -------------------------------------------------------
#if __has_builtin(__builtin_amdgcn_s_wait_asynccnt)
#define WAIT_ASYNC(n) __builtin_amdgcn_s_wait_asynccnt(n)
#else
#define WAIT_ASYNC(n) asm volatile("s_wait_asynccnt %0" ::"i"(n) : "memory")
#endif

// Async copy of 16 bytes global -> LDS (GLOBAL_LOAD_ASYNC_TO_LDS_B128, GV mode).
// Per ISA 15.18.3: copies 16 bytes from vaddr+offset into LDS at vdst+offset, tracked by ASYNCcnt.
__device__ __forceinline__ void async_copy_b128(unsigned lds_byte_addr,
                                                const float* gaddr) {
  asm volatile("global_load_async_to_lds_b128 %0, %1, off"
               :
               : "v"(lds_byte_addr), "v"(gaddr)
               : "memory");
}
__device__ __forceinline__ void async_copy_b128_off16(unsigned lds_byte_addr,
                                                      const float* gaddr) {
  asm volatile("global_load_async_to_lds_b128 %0, %1, off offset:16"
               :
               : "v"(lds_byte_addr), "v"(gaddr)
               : "memory");
}

// ---------------------------------------------------------------------------
// threefry2x32 (JAX PRNG) — reproduce jax.random.randint(key(seed), (), 0, N)
// ---------------------------------------------------------------------------
__device__ __forceinline__ unsigned rotl32(unsigned x, int r) {
  return (x << r) | (x >> (32 - r));
}

__device__ unsigned threefry2x32_first(unsigned k0, unsigned k1) {
  unsigned ks2 = 0x1BD11BDAu ^ k0 ^ k1;
  unsigned x0 = 0u, x1 = 0u;       // counter words (0 padded)
  x0 += k0; x1 += k1;
  const int RA[4] = {13, 15, 26, 6};
  const int RB[4] = {17, 29, 16, 24};
  unsigned ks[3] = {k0, k1, ks2};
  #pragma unroll
  for (int g = 0; g < 5; ++g) {
    const int* R = (g & 1) ? RB : RA;
    #pragma unroll
    for (int r = 0; r < 4; ++r) { x0 += x1; x1 = rotl32(x1, R[r]); x1 ^= x0; }
    unsigned i = (unsigned)(g + 1);
    x0 += ks[i % 3];
    x1 += ks[(i + 1) % 3] + i;
  }
  return x0;
}

// ---------------------------------------------------------------------------
// 1) Farthest point sampling: single workgroup, 1024 threads, sequential steps
// ---------------------------------------------------------------------------
__global__ void __launch_bounds__(1024)
fps_kernel(const float* __restrict__ pts, const int* __restrict__ rkey,
           int* __restrict__ sampled, float* __restrict__ dists) {
  __shared__ float rv[1024];
  __shared__ int   ri[1024];
  __shared__ float lpx, lpy, lpz;
  __shared__ int   lastIdx;
  const int t = threadIdx.x;

  for (int i = t; i < N_PTS; i += 1024) dists[i] = INFINITY;
  if (t == 0) {
    unsigned key = (unsigned)rkey[0];                 // jax.random.key(seed)
    unsigned bits = threefry2x32_first(0u, key);
    int first = (int)(bits & 0xFFFFu);                // % 65536
    sampled[0] = first;
    lastIdx = first;
  }
  __syncthreads();

  for (int s = 1; s < NSAMP; ++s) {
    if (t == 0) {
      const int li = lastIdx;
      lpx = pts[(size_t)li * 6 + 0];
      lpy = pts[(size_t)li * 6 + 1];
      lpz = pts[(size_t)li * 6 + 2];
    }
    __syncthreads();
    const int   li = lastIdx;
    const float bx = lpx, by = lpy, bz = lpz;
    float bv = -INFINITY; int bi = 0;
    for (int i = t; i < N_PTS; i += 1024) {
      const float dx = pts[(size_t)i * 6 + 0] - bx;
      const float dy = pts[(size_t)i * 6 + 1] - by;
      const float dz = pts[(size_t)i * 6 + 2] - bz;
      float nd = fminf(dists[i], dx * dx + dy * dy + dz * dz);
      if (i == li) nd = -INFINITY;
      dists[i] = nd;
      if (nd > bv) { bv = nd; bi = i; }
    }
    rv[t] = bv; ri[t] = bi;
    __syncthreads();
    for (int off = 512; off; off >>= 1) {
      if (t < off) {
        if (rv[t + off] > rv[t] ||
            (rv[t + off] == rv[t] && ri[t + off] < ri[t])) {
          rv[t] = rv[t + off]; ri[t] = ri[t + off];
        }
      }
      __syncthreads();
    }
    if (t == 0) { sampled[s] = ri[0]; lastIdx = ri[0]; }
    __syncthreads();
  }
}

// ---------------------------------------------------------------------------
// 2) squared norms of xyz
// ---------------------------------------------------------------------------
__global__ void pnorm_kernel(const float* __restrict__ pts, float* __restrict__ pnorm) {
  const int i = blockIdx.x * blockDim.x + threadIdx.x;
  if (i < N_PTS) {
    const float x = pts[(size_t)i * 6 + 0];
    const float y = pts[(size_t)i * 6 + 1];
    const float z = pts[(size_t)i * 6 + 2];
    pnorm[i] = x * x + y * y + z * z;
  }
}

// ---------------------------------------------------------------------------
// 3) approx 64-NN: 64 bins of 1024 points, argmin distance per bin
// ---------------------------------------------------------------------------
__global__ void __launch_bounds__(1024)
group_kernel(const float* __restrict__ pts, const int* __restrict__ sampled,
             const float* __restrict__ pnorm, int* __restrict__ groups) {
  __shared__ float cxs, cys, czs;
  __shared__ float rv[1024];
  __shared__ int   ri[1024];
  const int s = blockIdx.x;
  const int t = threadIdx.x;
  if (t == 0) {
    const int id = sampled[s];
    cxs = pts[(size_t)id * 6 + 0];
    cys = pts[(size_t)id * 6 + 1];
    czs = pts[(size_t)id * 6 + 2];
  }
  __syncthreads();
  const float cx = cxs, cy = cys, cz = czs;
  const int bin = t >> 4;
  const int sub = t & 15;
  float bestv = INFINITY; int besti = bin * 1024;
  const int base = bin * 1024 + sub;
  #pragma unroll 4
  for (int j = 0; j < 64; ++j) {
    const int p = base + j * 16;
    const float sc = pnorm[p] -
        2.0f * (cx * pts[(size_t)p * 6 + 0] +
                cy * pts[(size_t)p * 6 + 1] +
                cz * pts[(size_t)p * 6 + 2]);
    if (sc < bestv) { bestv = sc; besti = p; }
  }
  rv[t] = bestv; ri[t] = besti;
  __syncthreads();
  for (int off = 8; off; off >>= 1) {
    if (sub < off) {
      if (rv[t + off] < rv[t] ||
          (rv[t + off] == rv[t] && ri[t + off] < ri[t])) {
        rv[t] = rv[t + off]; ri[t] = ri[t + off];
      }
    }
    __syncthreads();
  }
  if (sub == 0) groups[s * NGRP + bin] = ri[t];
}

// ---------------------------------------------------------------------------
// 4) features = concat(cluster - centroid, cluster)  -> (131072, 12)
// ---------------------------------------------------------------------------
__global__ void feat_kernel(const float* __restrict__ pts, const int* __restrict__ sampled,
                            const int* __restrict__ groups, float* __restrict__ feat) {
  const int id = blockIdx.x * blockDim.x + threadIdx.x;   // 0..131071
  if (id >= NROWS) return;
  const int s  = id >> 6;
  const int j  = groups[id];
  const int ci = sampled[s];
  #pragma unroll
  for (int d = 0; d < 6; ++d) {
    const float cl = pts[(size_t)j  * 6 + d];
    const float ce = pts[(size_t)ci * 6 + d];
    feat[(size_t)id * FEAT_K + d]     = cl - ce;
    feat[(size_t)id * FEAT_K + 6 + d] = cl;
  }
}

// ---------------------------------------------------------------------------
// 5) GEMM1: (131072x12) @ (12x512) + b1  via V_WMMA_F32_16X16X4_F32
//    one 16x16 tile per wave, 8 waves per block
// ---------------------------------------------------------------------------
__global__ void __launch_bounds__(256)
gemm1_kernel(const float* __restrict__ feat, const float* __restrict__ W1,
             const float* __restrict__ b1, float* __restrict__ h1) {
  const int w    = threadIdx.x >> 5;
  const int lane = threadIdx.x & 31;
  const int lid  = lane & 15;
  const int half = lane >> 4;
  const int tile = blockIdx.x * 8 + w;       // 262144 tiles
  const int tm   = tile >> 5;                // N tiles = 512/16 = 32
  const int tn   = tile & 31;
  const int row  = tm * 16 + lid;
  const int col  = tn * 16 + lid;

  v8f acc = {};
  #pragma unroll
  for (int kk = 0; kk < FEAT_K; kk += 4) {
    const int ka = kk + 2 * half;
    v2f a = *(const v2f*)(feat + (size_t)row * FEAT_K + ka);   // 8B aligned
    v2f b;
    b.x = W1[(size_t)ka * EMB + col];
    b.y = W1[(size_t)(ka + 1) * EMB + col];
    acc = __builtin_amdgcn_wmma_f32_16x16x4_f32(false, a, false, b,
                                                (short)0, acc, false, false);
  }
  const float bias = b1[col];
  #pragma unroll
  for (int r = 0; r < 8; ++r) {
    const int gm = tm * 16 + r + 8 * half;
    h1[(size_t)gm * EMB + col] = acc[r] + bias;
  }
}

// ---------------------------------------------------------------------------
// 6) GEMM2: (131072x512) @ (512x512) + b2
//    LDS double-buffered via async global->LDS (ASYNCcnt), fp32 WMMA compute.
//    Block tile 128(M) x 64(N), 8 waves as 4x2, each wave 32x32 (2x2 WMMA).
// ---------------------------------------------------------------------------
__global__ void __launch_bounds__(256)
gemm2_kernel(const float* __restrict__ A, const float* __restrict__ W2,
             const float* __restrict__ b2, float* __restrict__ h2) {
  __shared__ float sA[2][128 * APITCH];
  __shared__ float sB[2][16 * BPITCH];

  const int t    = threadIdx.x;
  const int w    = t >> 5;
  const int lane = t & 31;
  const int lid  = lane & 15;
  const int half = lane >> 4;
  const int bm   = blockIdx.x >> 3;          // 1024 M blocks
  const int bn   = blockIdx.x & 7;           // 8 N blocks
  const int m0   = bm * 128;
  const int n0   = bn * 64;
  const int wm   = w & 3;                    // wave rows: wm*32
  const int wn   = w >> 2;                   // wave cols: wn*32

  v8f acc[2][2] = {};

  // staging assignment: A row t>>1, 8 floats at col (t&1)*8 (two B128s);
  //                     B row t>>4, 4 floats at col (t&15)*4 (one B128)
  const int la_row = t >> 1;
  const int la_c   = (t & 1) * 8;
  const int lb_row = t >> 4;
  const int lb_c   = (t & 15) * 4;

  const float* gA = A  + (size_t)(m0 + la_row) * EMB + la_c;   // + k0
  const float* gB = W2 + (size_t)lb_row * EMB + n0 + lb_c;     // + k0*EMB
  const unsigned ldsA0 = (unsigned)(size_t)&sA[0][la_row * APITCH + la_c];
  const unsigned ldsA1 = (unsigned)(size_t)&sA[1][la_row * APITCH + la_c];
  const unsigned ldsB0 = (unsigned)(size_t)&sB[0][lb_row * BPITCH + lb_c];
  const unsigned ldsB1 = (unsigned)(size_t)&sB[1][lb_row * BPITCH + lb_c];

  // prologue: fill buffer 0 with the k0 = 0 tiles
  async_copy_b128(ldsA0, gA);
  async_copy_b128_off16(ldsA0, gA);
  async_copy_b128(ldsB0, gB);

  int buf = 0;
  for (int k0 = 0; k0 < EMB; k0 += 16) {
    // everyone done computing from the buffer we are about to overwrite
    __syncthreads();
    const bool hasNext = (k0 + 16 < EMB);
    if (hasNext) {
      const float* ga = gA + (k0 + 16);
      const float* gb = gB + (size_t)(k0 + 16) * EMB;
      const unsigned la = (buf == 0) ? ldsA1 : ldsA0;
      const unsigned lb = (buf == 0) ? ldsB1 : ldsB0;
      async_copy_b128(la, ga);
      async_copy_b128_off16(la, ga);
      async_copy_b128(lb, gb);
      WAIT_ASYNC(3);    // 3 newest pending; current buffer's copies complete
    } else {
      WAIT_ASYNC(0);
    }
    __syncthreads();    // current buffer visible to all waves

    const float* __restrict__ cA = sA[buf];
    const float* __restrict__ cB = sB[buf];
    #pragma unroll
    for (int kk = 0; kk < 16; kk += 4) {
      const int ka = kk + 2 * half;
      v2f aF[2], bF[2];
      #pragma unroll
      for (int i = 0; i < 2; ++i) {
        const int ar = wm * 32 + i * 16 + lid;
        aF[i].x = cA[ar * APITCH + ka];
        aF[i].y = cA[ar * APITCH + ka + 1];
      }
      #pragma unroll
      for (int j = 0; j < 2; ++j) {
        const int bc = wn * 32 + j * 16 + lid;
        bF[j].x = cB[ka * BPITCH + bc];
        bF[j].y = cB[(ka + 1) * BPITCH + bc];
      }
      #pragma unroll
      for (int i = 0; i < 2; ++i)
        #pragma unroll
        for (int j = 0; j < 2; ++j)
          acc[i][j] = __builtin_amdgcn_wmma_f32_16x16x4_f32(
              false, aF[i], false, bF[j], (short)0, acc[i][j], false, false);
    }
    buf ^= 1;
  }

  #pragma unroll
  for (int i = 0; i < 2; ++i) {
    #pragma unroll
    for (int j = 0; j < 2; ++j) {
      const int gn   = n0 + wn * 32 + j * 16 + lid;
      const float bs = b2[gn];
      #pragma unroll
      for (int r = 0; r < 8; ++r) {
        const int gm = m0 + wm * 32 + i * 16 + r + 8 * half;
        h2[(size_t)gm * EMB + gn] = acc[i][j][r] + bs;
      }
    }
  }
}

// ---------------------------------------------------------------------------
// 7) BN statistics: per-channel sum / sumsq over 131072 rows
// ---------------------------------------------------------------------------
__global__ void zero_stats_kernel(float* __restrict__ sum, float* __restrict__ sumsq) {
  const int t = threadIdx.x;
  if (t < EMB) { sum[t] = 0.0f; sumsq[t] = 0.0f; }
}

__global__ void __launch_bounds__(256)
stats_kernel(const float* __restrict__ h, float* __restrict__ sum,
             float* __restrict__ sumsq) {
  const int b = blockIdx.x;          // 256 blocks, each 512 rows
  const int t = threadIdx.x;         // channels t and t+256
  float a0 = 0.f, q0 = 0.f, a1 = 0.f, q1 = 0.f;
  const size_t base = (size_t)b * 512 * EMB;
  for (int r = 0; r < 512; ++r) {
    const float v0 = h[base + (size_t)r * EMB + t];
    const float v1 = h[base + (size_t)r * EMB + t + 256];
    a0 += v0; q0 += v0 * v0;
    a1 += v1; q1 += v1 * v1;
  }
  atomicAdd(&sum[t], a0);        atomicAdd(&sumsq[t], q0);
  atomicAdd(&sum[t + 256], a1);  atomicAdd(&sumsq[t + 256], q1);
}

__global__ void finalize_stats_kernel(const float* __restrict__ sum,
                                      const float* __restrict__ sumsq,
                                      float* __restrict__ mean,
                                      float* __restrict__ rstd) {
  const int t = threadIdx.x;
  if (t < EMB) {
    const float inv_n = 1.0f / (float)NROWS;
    const float m = sum[t] * inv_n;
    const float v = sumsq[t] * inv_n - m * m;
    mean[t] = m;
    rstd[t] = rsqrtf(v + BN_EPS);
  }
}

// ---------------------------------------------------------------------------
// 8) BN1 + ReLU in place on h1 (float4 vectorized; 512 % 4 == 0)
// ---------------------------------------------------------------------------
__global__ void __launch_bounds__(256)
bn_relu_kernel(float* __restrict__ h, const float* __restrict__ scale,
               const float* __restrict__ bias, const float* __restrict__ mean,
               const float* __restrict__ rstd) {
  const size_t idx = (size_t)blockIdx.x * blockDim.x + threadIdx.x;  // float4 id
  const int c = (int)((idx * 4) & (EMB - 1));
  float4 v = ((float4*)h)[idx];
  v.x = fmaxf(0.f, scale[c + 0] * (v.x - mean[c + 0]) * rstd[c + 0] + bias[c + 0]);
  v.y = fmaxf(0.f, scale[c + 1] * (v.y - mean[c + 1]) * rstd[c + 1] + bias[c + 1]);
  v.z = fmaxf(0.f, scale[c + 2] * (v.z - mean[c + 2]) * rstd[c + 2] + bias[c + 2]);
  v.w = fmaxf(0.f, scale[c + 3] * (v.w - mean[c + 3]) * rstd[c + 3] + bias[c + 3]);
  ((float4*)h)[idx] = v;
}

// ---------------------------------------------------------------------------
// 9) BN2 + ReLU + max over the 64-group axis -> out (2048, 512)
// ---------------------------------------------------------------------------
__global__ void __launch_bounds__(256)
bn2_max_kernel(const float* __restrict__ h2, const float* __restrict__ scale,
               const float* __restrict__ bias, const float* __restrict__ mean,
               const float* __restrict__ rstd, float* __restrict__ out) {
  const int s = blockIdx.x;          // 2048
  const int t = threadIdx.x;         // channels t and t+256
  const int c0 = t, c1 = t + 256;
  const float sc0 = scale[c0], bi0 = bias[c0], me0 = mean[c0], rs0 = rstd[c0];
  const float sc1 = scale[c1], bi1 = bias[c1], me1 = mean[c1], rs1 = rstd[c1];
  float m0 = -INFINITY, m1 = -INFINITY;
  const size_t base = (size_t)s * NGRP * EMB;
  for (int g = 0; g < NGRP; ++g) {
    const float v0 = h2[base + (size_t)g * EMB + c0];
    const float v1 = h2[base + (size_t)g * EMB + c1];
    m0 = fmaxf(m0, fmaxf(0.f, sc0 * (v0 - me0) * rs0 + bi0));
    m1 = fmaxf(m1, fmaxf(0.f, sc1 * (v1 - me1) * rs1 + bi1));
  }
  out[(size_t)s * EMB + c0] = m0;
  out[(size_t)s * EMB + c1] = m1;
}

// ---------------------------------------------------------------------------
extern "C" void kernel_launch(void* const* d_in, const int* in_sizes, int n_in,
                              void* d_out, int out_size, void* d_ws, size_t ws_size,
                              hipStream_t stream) {
  const float* points = (const float*)d_in[0];
  const float* W1     = (const float*)d_in[1];
  const float* b1     = (const float*)d_in[2];
  const float* scale1 = (const float*)d_in[3];
  const float* bias1  = (const float*)d_in[4];
  const float* W2     = (const float*)d_in[5];
  const float* b2     = (const float*)d_in[6];
  const float* scale2 = (const float*)d_in[7];
  const float* bias2  = (const float*)d_in[8];
  const int*   rkey   = (const int*)d_in[9];
  float* out = (float*)d_out;

  char* ws = (char*)d_ws;
  int*   sampled = (int*)(ws + OFF_SAMPLED);
  float* dists   = (float*)(ws + OFF_DISTS);
  float* pnorm   = (float*)(ws + OFF_PNORM);
  int*   groups  = (int*)(ws + OFF_GROUPS);
  float* feat    = (float*)(ws + OFF_FEAT);
  float* ssum    = (float*)(ws + OFF_SUM);
  float* ssumsq  = (float*)(ws + OFF_SUMSQ);
  float* smean   = (float*)(ws + OFF_MEAN);
  float* srstd   = (float*)(ws + OFF_RSTD);
  float* h1      = (float*)(ws + OFF_H1);
  float* h2      = (float*)(ws + OFF_H2);

  // 1) FPS (sequential, one workgroup)
  fps_kernel<<<1, 1024, 0, stream>>>(points, rkey, sampled, dists);
  // 2) point xyz norms
  pnorm_kernel<<<N_PTS / 1024, 1024, 0, stream>>>(points, pnorm);
  // 3) approx top-64 grouping
  group_kernel<<<NSAMP, 1024, 0, stream>>>(points, sampled, pnorm, groups);
  // 4) feature construction (131072 x 12)
  feat_kernel<<<NROWS / 256, 256, 0, stream>>>(points, sampled, groups, feat);
  // 5) GEMM1 via WMMA  -> h1
  gemm1_kernel<<<(NROWS / 16) * (EMB / 16) / 8, 256, 0, stream>>>(feat, W1, b1, h1);
  // BN1 stats
  zero_stats_kernel<<<1, EMB, 0, stream>>>(ssum, ssumsq);
  stats_kernel<<<NROWS / 512, 256, 0, stream>>>(h1, ssum, ssumsq);
  finalize_stats_kernel<<<1, EMB, 0, stream>>>(ssum, ssumsq, smean, srstd);
  // BN1 + ReLU in place
  bn_relu_kernel<<<(int)(((size_t)NROWS * EMB / 4) / 256), 256, 0, stream>>>(
      h1, scale1, bias1, smean, srstd);
  // 6) GEMM2 via async-LDS double-buffered WMMA -> h2
  gemm2_kernel<<<(NROWS / 128) * (EMB / 64), 256, 0, stream>>>(h1, W2, b2, h2);
  // BN2 stats
  zero_stats_kernel<<<1, EMB, 0, stream>>>(ssum, ssumsq);
  stats_kernel<<<NROWS / 512, 256, 0, stream>>>(h2, ssum, ssumsq);
  finalize_stats_kernel<<<1, EMB, 0, stream>>>(ssum, ssumsq, smean, srstd);
  // BN2 + ReLU + group max-pool -> out (2048, 512)
  bn2_max_kernel<<<NSAMP, 256, 0, stream>>>(h2, scale2, bias2, smean, srstd, out);
}